// DNC_45191645888645
// MI455X (gfx1250) — compile-verified
//
#include <hip/hip_runtime.h>
#include <math.h>

// ---------------- problem constants ----------------
#define BB   16
#define SS   32
#define INP  256
#define HH   512
#define NN   512
#define WW   64
#define RR   4
#define OUTD 128
#define IFACE 471
#define CTRL 512          // IN + R*W
#define KTOT 1024         // CTRL + H
#define NG   2048         // 4*H
#define KOUT 768          // H + R*W
#define EPSV 1e-6f

// interface field offsets
#define OFF_RK   0     // R*W
#define OFF_RSTR 256   // R
#define OFF_WK   260   // W
#define OFF_WSTR 324   // 1
#define OFF_ER   325   // W
#define OFF_WV   389   // W
#define OFF_FG   453   // R
#define OFF_AG   457
#define OFF_WG   458
#define OFF_RM   459   // R*3

// ---------------- workspace layout (floats) ----------------
#define O_H       0u
#define O_C       8192u
#define O_MEM     16384u
#define O_USAGE   540672u
#define O_LINK    548864u
#define O_PREC    4743168u
#define O_READW   4751360u
#define O_WRITEW  4784128u
#define O_RWORDS  4792320u
#define STATE_END 4796416u
#define O_GATES   4796416u
#define O_IFACE   4829184u
#define O_CWL     4836720u
#define O_ALLOC   4844912u
#define O_WWSUM   4853104u
#define O_FWD     4853120u
#define O_BWD     4885888u
#define O_CR      4918656u

typedef __attribute__((ext_vector_type(2))) float v2f;
typedef __attribute__((ext_vector_type(8))) float v8f;

__device__ __forceinline__ float sigm(float x) { return 1.0f / (1.0f + __expf(-x)); }
__device__ __forceinline__ float oneplus(float x) {
  float sp = (x > 20.0f) ? x : log1pf(__expf(x));
  return 1.0f + sp;
}

// one WMMA K-segment: A rows contiguous (stride sA), B rows contiguous (stride sB)
__device__ __forceinline__ v8f wmma_seg_vv(v8f acc, const float* __restrict__ A, int sA,
                                           const float* __restrict__ B, int sB,
                                           int klen, int lm, int j, int half) {
  const float* ap = A + lm * sA + 2 * half;
  const float* bp = B + j * sB + 2 * half;
#pragma unroll 4
  for (int kk = 0; kk < klen; kk += 4) {
    v2f a  = *(const v2f*)(ap + kk);
    v2f bf = *(const v2f*)(bp + kk);
    acc = __builtin_amdgcn_wmma_f32_16x16x4_f32(false, a, false, bf, (short)0, acc, false, false);
  }
  return acc;
}

// one WMMA K-segment: A contiguous, B column-strided (B[k*ldB + j])
__device__ __forceinline__ v8f wmma_seg_vs(v8f acc, const float* __restrict__ A, int sA,
                                           const float* __restrict__ B, int ldB,
                                           int klen, int lm, int j, int half) {
  const float* ap = A + lm * sA + 2 * half;
  const float* bp = B + 2 * half * ldB + j;
#pragma unroll 4
  for (int kk = 0; kk < klen; kk += 4) {
    v2f a = *(const v2f*)(ap + kk);
    v2f bf;
    bf.x = bp[(size_t)kk * ldB];
    bf.y = bp[(size_t)kk * ldB + ldB];
    acc = __builtin_amdgcn_wmma_f32_16x16x4_f32(false, a, false, bf, (short)0, acc, false, false);
  }
  return acc;
}

// ---------------- zero state ----------------
__global__ void k_zero(float* ws) {
  unsigned g = blockIdx.x * blockDim.x + threadIdx.x;
  if (g < STATE_END) ws[g] = 0.0f;
}

// ---------------- LSTM gates GEMM: gates = ctrl_in @ w_ih^T + h @ w_hh^T + b ----------------
// grid 16 blocks x 256 threads; wave per 16-col tile; M=16 (batch), K=1024, N=2048
__global__ void k_gates(const float* __restrict__ x, const float* __restrict__ h,
                        const float* __restrict__ rwords,
                        const float* __restrict__ w_ih, const float* __restrict__ w_hh,
                        const float* __restrict__ b_ih, const float* __restrict__ b_hh,
                        float* __restrict__ gates, int t) {
  const int lane = threadIdx.x & 31, wave = threadIdx.x >> 5;
  const int half = lane >> 4, lm = lane & 15;
  const int j = (blockIdx.x * 8 + wave) * 16 + lm;
  const float* xt = x + (size_t)t * BB * INP;
  v8f acc = {};
  // K segment 1: x_t (k 0..255) against w_ih[:, 0:256]
  acc = wmma_seg_vv(acc, xt,     INP, w_ih,       CTRL, INP, lm, j, half);
  // K segment 2: read_words (k 256..511) against w_ih[:, 256:512]
  acc = wmma_seg_vv(acc, rwords, 256, w_ih + INP, CTRL, 256, lm, j, half);
  // K segment 3: h (k 512..1023) against w_hh
  acc = wmma_seg_vv(acc, h,      HH,  w_hh,       HH,   HH,  lm, j, half);
  float bias = b_ih[j] + b_hh[j];
#pragma unroll
  for (int v = 0; v < 8; ++v) {
    int m = v + 8 * half;
    gates[m * NG + j] = acc[v] + bias;
  }
}

// ---------------- LSTM pointwise ----------------
__global__ void k_lstm(const float* __restrict__ gates, float* __restrict__ c, float* __restrict__ h) {
  int tid = blockIdx.x * blockDim.x + threadIdx.x; // B*H
  int b = tid >> 9, i = tid & 511;
  const float* g = gates + b * NG;
  float gi = g[i], gf = g[HH + i], gg = g[2 * HH + i], go = g[3 * HH + i];
  float cn = sigm(gf) * c[tid] + sigm(gi) * tanhf(gg);
  c[tid] = cn;
  h[tid] = sigm(go) * tanhf(cn);
}

// ---------------- interface GEMM: iface = h @ W_iface + b ----------------
// grid 4 blocks x 256; 30 active tiles (N=471 padded to 480); K=512
__global__ void k_iface(const float* __restrict__ h, const float* __restrict__ Wif,
                        const float* __restrict__ bif, float* __restrict__ iface) {
  const int lane = threadIdx.x & 31, wave = threadIdx.x >> 5;
  const int tile = blockIdx.x * 8 + wave;
  if (tile >= 30) return; // wave-uniform
  const int half = lane >> 4, lm = lane & 15;
  const int j = tile * 16 + lm;
  const int jc = (j < IFACE) ? j : (IFACE - 1);   // clamp: loads unconditional
  v8f acc = wmma_seg_vs((v8f){}, h, HH, Wif, IFACE, HH, lm, jc, half);
  if (j < IFACE) {
    float bias = bif[j];
#pragma unroll
    for (int v = 0; v < 8; ++v) {
      int m = v + 8 * half;
      iface[m * IFACE + j] = acc[v] + bias;
    }
  }
}

// ---------------- usage update + write-content logits ----------------
__global__ void k_usage_cw(const float* __restrict__ iface, const float* __restrict__ read_w,
                           const float* __restrict__ write_w, float* __restrict__ usage,
                           const float* __restrict__ mem, float* __restrict__ cwl) {
  int tid = blockIdx.x * blockDim.x + threadIdx.x; // B*N
  int b = tid >> 9, n = tid & 511;
  const float* ifb = iface + b * IFACE;
  float psi = 1.0f;
#pragma unroll
  for (int r = 0; r < RR; ++r) {
    float fg = sigm(ifb[OFF_FG + r]);
    psi *= 1.0f - fg * read_w[(b * RR + r) * NN + n];
  }
  float u = usage[tid], wwo = write_w[tid];
  u = (u + wwo - u * wwo) * psi;
  usage[tid] = u;
  const float* mrow = mem + (size_t)(b * NN + n) * WW;
  float dot = 0, nm = 0, nk = 0;
#pragma unroll 8
  for (int w = 0; w < WW; ++w) {
    float mv = mrow[w], kv = ifb[OFF_WK + w];
    dot += mv * kv; nm += mv * mv; nk += kv * kv;
  }
  float cosv = dot / (sqrtf(nk) * sqrtf(nm) + EPSV);
  cwl[tid] = oneplus(ifb[OFF_WSTR]) * cosv;
}

// ---------------- in-place softmax over rows of 512 ----------------
__global__ void k_softmax512(float* __restrict__ d) {
  __shared__ float s[256];
  int tid = threadIdx.x;
  float* row = d + blockIdx.x * 512;
  float a0 = row[tid], a1 = row[tid + 256];
  s[tid] = fmaxf(a0, a1); __syncthreads();
  for (int st = 128; st > 0; st >>= 1) { if (tid < st) s[tid] = fmaxf(s[tid], s[tid + st]); __syncthreads(); }
  float M = s[0]; __syncthreads();
  float e0 = __expf(a0 - M), e1 = __expf(a1 - M);
  s[tid] = e0 + e1; __syncthreads();
  for (int st = 128; st > 0; st >>= 1) { if (tid < st) s[tid] += s[tid + st]; __syncthreads(); }
  float inv = 1.0f / s[0];
  row[tid] = e0 * inv; row[tid + 256] = e1 * inv;
}

// ---------------- allocation: stable bitonic argsort + cumprod scan ----------------
__global__ void k_alloc(const float* __restrict__ usage, float* __restrict__ alloc) {
  __shared__ float skey[512];
  __shared__ int   sidx[512];
  __shared__ float sp[512];
  int b = blockIdx.x, tid = threadIdx.x;
  skey[tid] = EPSV + (1.0f - EPSV) * usage[b * NN + tid];
  sidx[tid] = tid;
  __syncthreads();
  for (int k = 2; k <= 512; k <<= 1) {
    for (int jj = k >> 1; jj > 0; jj >>= 1) {
      int ixj = tid ^ jj;
      if (ixj > tid) {
        bool up = ((tid & k) == 0);
        float ka = skey[tid], kb = skey[ixj];
        int ia = sidx[tid], ib = sidx[ixj];
        bool gt = (ka > kb) || (ka == kb && ia > ib);
        bool sw = up ? gt : !gt;
        if (sw) { skey[tid] = kb; skey[ixj] = ka; sidx[tid] = ib; sidx[ixj] = ia; }
      }
      __syncthreads();
    }
  }
  // exclusive cumulative product of sorted u
  sp[tid] = (tid == 0) ? 1.0f : skey[tid - 1];
  __syncthreads();
  for (int off = 1; off < 512; off <<= 1) {
    float v = (tid >= off) ? sp[tid - off] : 1.0f;
    __syncthreads();
    sp[tid] *= v;
    __syncthreads();
  }
  alloc[b * NN + sidx[tid]] = (1.0f - skey[tid]) * sp[tid];
}

// ---------------- write weights + sum reduction ----------------
__global__ void k_writew(const float* __restrict__ iface, const float* __restrict__ alloc,
                         const float* __restrict__ cw, float* __restrict__ write_w,
                         float* __restrict__ wwsum) {
  __shared__ float s[512];
  int b = blockIdx.x, n = threadIdx.x;
  const float* ifb = iface + b * IFACE;
  float ag = sigm(ifb[OFF_AG]), wg = sigm(ifb[OFF_WG]);
  float ww = wg * (ag * alloc[b * NN + n] + (1.0f - ag) * cw[b * NN + n]);
  write_w[b * NN + n] = ww;
  s[n] = ww; __syncthreads();
  for (int st = 256; st > 0; st >>= 1) { if (n < st) s[n] += s[n + st]; __syncthreads(); }
  if (n == 0) wwsum[b] = s[0];
}

// ---------------- memory erase/write ----------------
__global__ void k_mem(const float* __restrict__ iface, const float* __restrict__ write_w,
                      float* __restrict__ mem) {
  int g = blockIdx.x * blockDim.x + threadIdx.x; // B*N*W
  int b = g >> 15, rem = g & 32767, n = rem >> 6, w = rem & 63;
  const float* ifb = iface + b * IFACE;
  float ww = write_w[b * NN + n];
  float e = sigm(ifb[OFF_ER + w]), v = sigm(ifb[OFF_WV + w]);
  mem[g] = mem[g] * (1.0f - ww * e) + ww * v;
}

// ---------------- link update ----------------
__global__ void k_link(float* __restrict__ link, const float* __restrict__ write_w,
                       const float* __restrict__ prec) {
  unsigned g = blockIdx.x * blockDim.x + threadIdx.x; // B*N*N
  int b = g >> 18, rem = g & 262143, i = rem >> 9, jj = rem & 511;
  float wwi = write_w[b * NN + i], wwj = write_w[b * NN + jj], po = prec[b * NN + jj];
  float L = link[g];
  L = (1.0f - wwi - wwj) * L + wwi * po;
  link[g] = (i == jj) ? 0.0f : L;
}

// ---------------- fwd/bwd link walks ----------------
__global__ void k_fwdbwd(const float* __restrict__ link, const float* __restrict__ read_w,
                         float* __restrict__ fwd, float* __restrict__ bwd) {
  int b = blockIdx.x >> 3, itile = blockIdx.x & 7;
  int r = threadIdx.x >> 6, il = threadIdx.x & 63;
  int i = itile * 64 + il;
  const float* Lb = link + (size_t)b * NN * NN;
  const float* rw = read_w + (b * RR + r) * NN;
  float f = 0.0f, bd = 0.0f;
  for (int jj = 0; jj < NN; ++jj) {
    float rv = rw[jj];
    f  += Lb[(size_t)i * NN + jj] * rv;   // row stream
    bd += Lb[(size_t)jj * NN + i] * rv;   // column, coalesced over i
  }
  fwd[(b * RR + r) * NN + i] = f;
  bwd[(b * RR + r) * NN + i] = bd;
}

// ---------------- precedence update (after link) ----------------
__global__ void k_prec(float* __restrict__ prec, const float* __restrict__ wwsum,
                       const float* __restrict__ write_w) {
  int tid = blockIdx.x * blockDim.x + threadIdx.x; // B*N
  int b = tid >> 9;
  prec[tid] = (1.0f - wwsum[b]) * prec[tid] + write_w[tid];
}

// ---------------- read content logits ----------------
__global__ void k_crl(const float* __restrict__ iface, const float* __restrict__ mem,
                      float* __restrict__ crl) {
  int g = blockIdx.x * blockDim.x + threadIdx.x; // B*R*N
  int b = g >> 11, rem = g & 2047, r = rem >> 9, n = rem & 511;
  const float* ifb = iface + b * IFACE;
  const float* key = ifb + OFF_RK + r * WW;
  const float* mrow = mem + (size_t)(b * NN + n) * WW;
  float dot = 0, nm = 0, nk = 0;
#pragma unroll 8
  for (int w = 0; w < WW; ++w) {
    float mv = mrow[w], kv = key[w];
    dot += mv * kv; nm += mv * mv; nk += kv * kv;
  }
  float cosv = dot / (sqrtf(nk) * sqrtf(nm) + EPSV);
  crl[g] = oneplus(ifb[OFF_RSTR + r]) * cosv;
}

// ---------------- read weight mix ----------------
__global__ void k_readw(const float* __restrict__ iface, const float* __restrict__ bwd,
                        const float* __restrict__ cr, const float* __restrict__ fwd,
                        float* __restrict__ read_w) {
  int g = blockIdx.x * blockDim.x + threadIdx.x; // B*R*N
  int b = g >> 11, rem = g & 2047, r = rem >> 9;
  const float* rm = iface + b * IFACE + OFF_RM + r * 3;
  float x0 = rm[0], x1 = rm[1], x2 = rm[2];
  float mx = fmaxf(x0, fmaxf(x1, x2));
  float e0 = __expf(x0 - mx), e1 = __expf(x1 - mx), e2 = __expf(x2 - mx);
  float inv = 1.0f / (e0 + e1 + e2);
  read_w[g] = (e0 * bwd[g] + e1 * cr[g] + e2 * fwd[g]) * inv;
}

// ---------------- read words ----------------
__global__ void k_rwords(const float* __restrict__ read_w, const float* __restrict__ mem,
                         float* __restrict__ rwords) {
  int b = blockIdx.x >> 2, r = blockIdx.x & 3, w = threadIdx.x;
  const float* rw = read_w + (b * RR + r) * NN;
  const float* mb = mem + (size_t)b * NN * WW;
  float acc = 0.0f;
  for (int n = 0; n < NN; ++n) acc += rw[n] * mb[n * WW + w];
  rwords[(b * RR + r) * WW + w] = acc;
}

// ---------------- output GEMM: out = [h, read_words] @ W_out + b_out ----------------
// 1 block x 256; 8 tiles of 16 cols; K=768
__global__ void k_out(const float* __restrict__ h, const float* __restrict__ rwords,
                      const float* __restrict__ Wout, const float* __restrict__ bout,
                      float* __restrict__ out, int t) {
  const int lane = threadIdx.x & 31, wave = threadIdx.x >> 5;
  const int half = lane >> 4, lm = lane & 15;
  const int j = wave * 16 + lm;
  v8f acc = {};
  // K segment 1: h (k 0..511)
  acc = wmma_seg_vs(acc, h, HH, Wout, OUTD, HH, lm, j, half);
  // K segment 2: read_words (k 512..767)
  acc = wmma_seg_vs(acc, rwords, 256, Wout + (size_t)HH * OUTD, OUTD, 256, lm, j, half);
  float bias = bout[j];
#pragma unroll
  for (int v = 0; v < 8; ++v) {
    int m = v + 8 * half;
    out[(size_t)t * BB * OUTD + m * OUTD + j] = acc[v] + bias;
  }
}

// ---------------- host launcher ----------------
extern "C" void kernel_launch(void* const* d_in, const int* in_sizes, int n_in,
                              void* d_out, int out_size, void* d_ws, size_t ws_size,
                              hipStream_t stream) {
  const float* inputs = (const float*)d_in[0];
  const float* w_ih   = (const float*)d_in[1];
  const float* w_hh   = (const float*)d_in[2];
  const float* b_ih   = (const float*)d_in[3];
  const float* b_hh   = (const float*)d_in[4];
  const float* W_if   = (const float*)d_in[5];
  const float* b_if   = (const float*)d_in[6];
  const float* W_out  = (const float*)d_in[7];
  const float* b_out  = (const float*)d_in[8];
  float* out = (float*)d_out;
  float* ws  = (float*)d_ws;

  float* h      = ws + O_H;
  float* c      = ws + O_C;
  float* mem    = ws + O_MEM;
  float* usage  = ws + O_USAGE;
  float* link   = ws + O_LINK;
  float* prec   = ws + O_PREC;
  float* read_w = ws + O_READW;
  float* write_w= ws + O_WRITEW;
  float* rwords = ws + O_RWORDS;
  float* gates  = ws + O_GATES;
  float* iface  = ws + O_IFACE;
  float* cwl    = ws + O_CWL;
  float* alloc  = ws + O_ALLOC;
  float* wwsum  = ws + O_WWSUM;
  float* fwd    = ws + O_FWD;
  float* bwd    = ws + O_BWD;
  float* cr     = ws + O_CR;

  k_zero<<<STATE_END / 256, 256, 0, stream>>>(ws);

  for (int t = 0; t < SS; ++t) {
    k_gates<<<16, 256, 0, stream>>>(inputs, h, rwords, w_ih, w_hh, b_ih, b_hh, gates, t);
    k_lstm<<<32, 256, 0, stream>>>(gates, c, h);
    k_iface<<<4, 256, 0, stream>>>(h, W_if, b_if, iface);
    k_usage_cw<<<32, 256, 0, stream>>>(iface, read_w, write_w, usage, mem, cwl);
    k_softmax512<<<BB, 256, 0, stream>>>(cwl);
    k_alloc<<<BB, 512, 0, stream>>>(usage, alloc);
    k_writew<<<BB, 512, 0, stream>>>(iface, alloc, cwl, write_w, wwsum);
    k_mem<<<2048, 256, 0, stream>>>(iface, write_w, mem);
    k_link<<<16384, 256, 0, stream>>>(link, write_w, prec);
    k_fwdbwd<<<128, 256, 0, stream>>>(link, read_w, fwd, bwd);
    k_prec<<<32, 256, 0, stream>>>(prec, wwsum, write_w);
    k_crl<<<128, 256, 0, stream>>>(iface, mem, cr);
    k_softmax512<<<BB * RR, 256, 0, stream>>>(cr);
    k_readw<<<128, 256, 0, stream>>>(iface, bwd, cr, fwd, read_w);
    k_rwords<<<BB * RR, 64, 0, stream>>>(read_w, mem, rwords);
    k_out<<<1, 256, 0, stream>>>(h, rwords, W_out, b_out, out, t);
  }
}